// FusedLocalNormalizedCrossCorrelationLoss_34222299415109
// MI455X (gfx1250) — compile-verified
//
#include <hip/hip_runtime.h>

// ---------------------------------------------------------------------------
// Fused Local Normalized Cross-Correlation loss for [2,1,160,160,160] f32.
// Separable 5-tap box filters expressed as banded-matrix GEMMs using
// V_WMMA_F32_16X16X4_F32 (5 chained K=4 WMMAs realize a 16x20 x 20x16 matmul).
// Single pass over HBM inputs (~65.5MB, L2-resident) -> partial sums -> scalar.
// Deterministic: fixed tiling, no floating-point atomics.
// Stage A computes all 5 channels per line-group in one wave-task: uniform
// control flow around WMMA (EXEC all-ones), iv/jv loaded once per K-chunk.
// ---------------------------------------------------------------------------

typedef float v2f __attribute__((ext_vector_type(2)));
typedef float v4f __attribute__((ext_vector_type(4)));
typedef float v8f __attribute__((ext_vector_type(8)));

#define SZ   160          // cube side
#define TD   4            // output depth per block
#define DP   (TD + 4)     // 8 input d-planes (halo 2 each side)
#define HP   20           // input h span (16 + 4 halo)
#define WP   20           // input w span
#define NLINES (DP * HP)  // 160 (d,h) lines for the W-pass
#define NGROUPS (NLINES / 16) // 10 WMMA line-groups
#define NTD  (SZ / TD)    // 40
#define NTH  (SZ / 16)    // 10
#define NTW  (SZ / 16)    // 10
#define NBLK (2 * NTD * NTH * NTW) // 8000 blocks
#define INV_N (1.0f / 8192000.0f)  // 1 / (2*160^3)

#define WMMA_F32_4(a, b, c) \
    __builtin_amdgcn_wmma_f32_16x16x4_f32(false, (a), false, (b), (short)0, (c), false, false)

__global__ __launch_bounds__(256)
void lncc_tile_kernel(const float* __restrict__ I,
                      const float* __restrict__ J,
                      float* __restrict__ partial)
{
    __shared__ __align__(16) float sI[DP * HP * WP];      // 3200 f  (12.8 KB)
    __shared__ __align__(16) float sJ[DP * HP * WP];      // 3200 f  (12.8 KB)
    __shared__ __align__(16) float sA[5 * NLINES * 16];   // 12800 f (51.2 KB) W-filtered
    __shared__ __align__(16) float sB[5 * DP * 16 * 16];  // 10240 f (41.0 KB) HW-filtered
    __shared__ __align__(16) float red[256];

    const int tid  = threadIdx.x;
    const int lane = tid & 31;
    const int wave = tid >> 5;
    const int n16  = lane & 15;          // N index (line / column)
    const bool hi  = lane >= 16;
    const int klo  = hi ? 2 : 0;         // K lane split for 32-bit A/B operands
    const int moff = hi ? 8 : 0;         // M offset for C/D rows

    // tile coordinates
    int bid = blockIdx.x;
    const int wt = bid % NTW; bid /= NTW;
    const int ht = bid % NTH; bid /= NTH;
    const int dt = bid % NTD; bid /= NTD;
    const int b  = bid;                                  // batch 0..1
    const int d0 = dt * TD, h0 = ht * 16, w0 = wt * 16;
    const int base = b * SZ * SZ * SZ;

    // ---- load zero-padded halo tiles (SAME conv semantics) ----------------
    for (int idx = tid; idx < DP * HP * WP; idx += 256) {
        const int ww = idx % WP;
        const int hh = (idx / WP) % HP;
        const int dd = idx / (WP * HP);
        const int gd = d0 + dd - 2, gh = h0 + hh - 2, gw = w0 + ww - 2;
        float iv = 0.0f, jv = 0.0f;
        if ((unsigned)gd < SZ && (unsigned)gh < SZ && (unsigned)gw < SZ) {
            const int g = base + (gd * SZ + gh) * SZ + gw;
            iv = I[g];
            jv = J[g];
        }
        sI[idx] = iv;
        sJ[idx] = jv;
    }
    __syncthreads();

    // Banded mean-filter A-operand for K-chunk c:
    // A[m][k] = 0.2 iff m <= k <= m+4 (out m maps to halo'd inputs m..m+4).
    // 16x4 f32 A layout: lane L holds M=L%16; lanes<16 K={4c,4c+1},
    // lanes>=16 K={4c+2,4c+3} in vgpr0/vgpr1.
    auto filtA = [&](int c) -> v2f {
        const int k0 = 4 * c + klo;
        v2f a;
        a.x = (k0     >= n16 && k0     <= n16 + 4) ? 0.2f : 0.0f;
        a.y = (k0 + 1 >= n16 && k0 + 1 <= n16 + 4) ? 0.2f : 0.0f;
        return a;
    };

    // ---- Stage A: W-axis filter, all 5 channels per line-group ------------
    // OUT(16 w_out x 16 lines) = A(16x20 banded) * DATA(20 w_in x 16 lines)
    // lines = flattened (d,h); 10 groups over 8 waves; 25 WMMAs per group.
    for (int g = wave; g < NGROUPS; g += 8) {
        const int l = g * 16 + n16;                  // this lane's line (d*HP+h)
        v8f aIJ = {}, aII = {}, aJJ = {}, aI = {}, aJ = {};
#pragma unroll
        for (int c = 0; c < 5; ++c) {
            const int w = 4 * c + klo;               // even -> 8B aligned b64 load
            const v2f fa = filtA(c);
            const v2f iv = *(const v2f*)&sI[l * WP + w];
            const v2f jv = *(const v2f*)&sJ[l * WP + w];
            aIJ = WMMA_F32_4(fa, iv * jv, aIJ);
            aII = WMMA_F32_4(fa, iv * iv, aII);
            aJJ = WMMA_F32_4(fa, jv * jv, aJJ);
            aI  = WMMA_F32_4(fa, iv,      aI);
            aJ  = WMMA_F32_4(fa, jv,      aJ);
        }
        // D layout: vgpr r -> w_out = r + moff, column n16 = line l
        float* d0p = &sA[(0 * NLINES + l) * 16 + moff];
        float* d1p = &sA[(1 * NLINES + l) * 16 + moff];
        float* d2p = &sA[(2 * NLINES + l) * 16 + moff];
        float* d3p = &sA[(3 * NLINES + l) * 16 + moff];
        float* d4p = &sA[(4 * NLINES + l) * 16 + moff];
        *(v4f*)(d0p) = aIJ.lo; *(v4f*)(d0p + 4) = aIJ.hi;
        *(v4f*)(d1p) = aII.lo; *(v4f*)(d1p + 4) = aII.hi;
        *(v4f*)(d2p) = aJJ.lo; *(v4f*)(d2p + 4) = aJJ.hi;
        *(v4f*)(d3p) = aI.lo;  *(v4f*)(d3p + 4) = aI.hi;
        *(v4f*)(d4p) = aJ.lo;  *(v4f*)(d4p + 4) = aJ.hi;
    }
    __syncthreads();

    // ---- Stage B: H-axis filter via WMMA ----------------------------------
    // OUT(16 h_out x 16 w) = A(16x20 banded) * sA(20 h_in x 16 w), per (ch,d).
    // 40 tasks over 8 waves; DP=8 is a power of two -> cheap task split.
    for (int task = wave; task < 5 * DP; task += 8) {
        const int d  = task & (DP - 1);
        const int ch = task >> 3;
        const int rowbase = (ch * NLINES + d * HP) * 16 + n16;
        v8f acc = {};
#pragma unroll
        for (int c = 0; c < 5; ++c) {
            const int k0 = 4 * c + klo;              // h_in
            v2f bv;
            bv.x = sA[rowbase + (k0    ) * 16];
            bv.y = sA[rowbase + (k0 + 1) * 16];
            acc = WMMA_F32_4(filtA(c), bv, acc);
        }
#pragma unroll
        for (int r = 0; r < 8; ++r)                  // h_out = r + moff, w = n16
            sB[((ch * DP + d) * 16 + (r + moff)) * 16 + n16] = acc[r];
    }
    __syncthreads();

    // ---- Stage C: D-axis filter + NCC epilogue (cheap VALU) ---------------
    float local = 0.0f;
#pragma unroll
    for (int t = 0; t < TD; ++t) {
        const int v = tid + (t << 8);                // 256 threads x 4 voxels
        const int w = v & 15;
        const int h = (v >> 4) & 15;
        const int tt = v >> 8;
        float m[5];
#pragma unroll
        for (int ch = 0; ch < 5; ++ch) {
            float s = 0.0f;
#pragma unroll
            for (int k = 0; k < 5; ++k)
                s += sB[((ch * DP + (tt + k)) * 16 + h) * 16 + w];
            m[ch] = 0.2f * s;
        }
        const float cross = m[0] - m[3] * m[4];
        const float var_i = m[1] - m[3] * m[3];
        const float var_j = m[2] - m[4] * m[4];
        local += (cross * cross) / (var_i * var_j + 1e-5f);
    }

    // block reduction (fixed order -> deterministic)
    red[tid] = local;
    __syncthreads();
    for (int s = 128; s > 0; s >>= 1) {
        if (tid < s) red[tid] += red[tid + s];
        __syncthreads();
    }
    if (tid == 0) partial[blockIdx.x] = red[0];
}

__global__ __launch_bounds__(256)
void lncc_reduce_kernel(const float* __restrict__ partial, float* __restrict__ out)
{
    __shared__ float red[256];
    float local = 0.0f;
    for (int i = threadIdx.x; i < NBLK; i += 256) local += partial[i];
    red[threadIdx.x] = local;
    __syncthreads();
    for (int s = 128; s > 0; s >>= 1) {
        if (threadIdx.x < s) red[threadIdx.x] += red[threadIdx.x + s];
        __syncthreads();
    }
    if (threadIdx.x == 0) out[0] = -red[0] * INV_N;   // -mean(ncc)
}

extern "C" void kernel_launch(void* const* d_in, const int* in_sizes, int n_in,
                              void* d_out, int out_size, void* d_ws, size_t ws_size,
                              hipStream_t stream)
{
    const float* I = (const float*)d_in[0];
    const float* J = (const float*)d_in[1];
    float* out     = (float*)d_out;
    float* partial = (float*)d_ws;     // NBLK floats = 32 KB of scratch

    lncc_tile_kernel<<<NBLK, 256, 0, stream>>>(I, J, partial);
    lncc_reduce_kernel<<<1, 256, 0, stream>>>(partial, out);
}